// GDMSR_LightGCN_72945724555838
// MI455X (gfx1250) — compile-verified
//
#include <hip/hip_runtime.h>
#include <hip/hip_bf16.h>

#define DIM 64

typedef __attribute__((ext_vector_type(2))) float v2f;
typedef __attribute__((ext_vector_type(8))) float v8f;

// ---------------------------------------------------------------------------
// Degree accumulation: degree[r] += 1 for both directions of each edge.
// 4M float atomics, all L2-resident (1.2 MB table).
// ---------------------------------------------------------------------------
__global__ void k_degree(const int* __restrict__ eu, const int* __restrict__ ei,
                         float* __restrict__ deg, int E, int n_user) {
    int e = blockIdx.x * blockDim.x + threadIdx.x;
    if (e >= E) return;
    atomicAdd(&deg[eu[e]], 1.0f);
    atomicAdd(&deg[n_user + ei[e]], 1.0f);
}

// d_inv_sqrt = (degree + 1e-8)^-0.5, in place.
__global__ void k_rsqrt(float* __restrict__ deg, int n) {
    int i = blockIdx.x * blockDim.x + threadIdx.x;
    if (i < n) deg[i] = rsqrtf(deg[i] + 1e-8f);
}

// ---------------------------------------------------------------------------
// SpMM layer: dst[u] += nm*src[v]; dst[v] += nm*src[u] for every edge (u,v),
// nm = dinv[u]*dinv[v]. 16 threads per edge, float4 per thread (64 floats/row).
// src/dst are both L2-resident (76.8 MB each vs 192 MB L2), so gathers and
// global_atomic_add_f32 scatters stay on-chip.
// ---------------------------------------------------------------------------
__global__ void k_spmm(const int* __restrict__ eu, const int* __restrict__ ei,
                       const float* __restrict__ dinv, const float* __restrict__ src,
                       float* __restrict__ dst, int E, int n_user) {
    long long t = (long long)blockIdx.x * blockDim.x + threadIdx.x;
    int e    = (int)(t >> 4);
    int lane = (int)(t & 15);
    if (e >= E) return;
    int u = eu[e];
    int v = n_user + ei[e];
    float nm = dinv[u] * dinv[v];
    const float4* su = (const float4*)(src + (size_t)u * DIM);
    const float4* sv = (const float4*)(src + (size_t)v * DIM);
    float4 a = su[lane];   // src[u] chunk
    float4 b = sv[lane];   // src[v] chunk
    float* du = dst + (size_t)u * DIM + lane * 4;
    float* dv = dst + (size_t)v * DIM + lane * 4;
    atomicAdd(du + 0, nm * b.x); atomicAdd(du + 1, nm * b.y);
    atomicAdd(du + 2, nm * b.z); atomicAdd(du + 3, nm * b.w);
    atomicAdd(dv + 0, nm * a.x); atomicAdd(dv + 1, nm * a.y);
    atomicAdd(dv + 2, nm * a.z); atomicAdd(dv + 3, nm * a.w);
}

// ---------------------------------------------------------------------------
// Initialize compact per-query accumulators with layer-0 embeddings.
// ---------------------------------------------------------------------------
__global__ void k_init_acc(const int* __restrict__ uid, const int* __restrict__ iid,
                           const float* __restrict__ uemb, const float* __restrict__ iemb,
                           float* __restrict__ uacc, float* __restrict__ iacc, int B) {
    long long t = (long long)blockIdx.x * blockDim.x + threadIdx.x;
    int b    = (int)(t >> 4);
    int lane = (int)(t & 15);
    if (b >= B) return;
    ((float4*)(uacc + (size_t)b * DIM))[lane] =
        ((const float4*)(uemb + (size_t)uid[b] * DIM))[lane];
    ((float4*)(iacc + (size_t)b * DIM))[lane] =
        ((const float4*)(iemb + (size_t)iid[b] * DIM))[lane];
}

// After each layer, add emb[user_ids[b]] / emb[n_user+item_ids[b]] into the
// compact accumulators (avoids a third full 76.8 MB acc buffer).
__global__ void k_acc(const int* __restrict__ uid, const int* __restrict__ iid,
                      const float* __restrict__ emb,
                      float* __restrict__ uacc, float* __restrict__ iacc,
                      int B, int n_user) {
    long long t = (long long)blockIdx.x * blockDim.x + threadIdx.x;
    int b    = (int)(t >> 4);
    int lane = (int)(t & 15);
    if (b >= B) return;
    float4 gu = ((const float4*)(emb + (size_t)uid[b] * DIM))[lane];
    float4 gi = ((const float4*)(emb + (size_t)(n_user + iid[b]) * DIM))[lane];
    float4* pu = (float4*)(uacc + (size_t)b * DIM) + lane;
    float4* pi = (float4*)(iacc + (size_t)b * DIM) + lane;
    float4 cu = *pu, ci = *pi;
    cu.x += gu.x; cu.y += gu.y; cu.z += gu.z; cu.w += gu.w;
    ci.x += gi.x; ci.y += gi.y; ci.z += gi.z; ci.w += gi.w;
    *pu = cu; *pi = ci;
}

// ---------------------------------------------------------------------------
// Final predictions via V_WMMA_F32_16X16X4_F32.
// One wave32 per tile of 16 queries: C = U * I^T accumulated over K=64 in 16
// k=4 steps; predictions are diag(C) * 0.25.
// A layout (32-bit A 16x4): lanes 0-15 -> M=lane, VGPRs hold K=k0,k0+1;
// lanes 16-31 -> M=lane-16, K=k0+2,k0+3. B (=I^T) mirrors this per-lane.
// C/D layout (32-bit 16x16): VGPR v, lanes 0-15 -> M=v, N=lane;
// lanes 16-31 -> M=v+8, N=lane-16.  So diag element m is:
//   m<8  : VGPR m   @ lane m
//   m>=8 : VGPR m-8 @ lane m+8
// EXEC stays all-ones through the WMMA chain (row clamp instead of return).
// ---------------------------------------------------------------------------
__global__ void k_predict(const float* __restrict__ uacc, const float* __restrict__ iacc,
                          float* __restrict__ out, int B) {
    int tile = blockIdx.x;
    int lane = threadIdx.x;            // 0..31 (wave32)
    int row  = lane & 15;
    int kh   = (lane >> 4) * 2;        // 0 for lanes 0-15, 2 for lanes 16-31
    int r = tile * 16 + row;
    if (r >= B) r = B - 1;             // clamp; keeps EXEC all-ones
    const float* up = uacc + (size_t)r * DIM;
    const float* ip = iacc + (size_t)r * DIM;

    v8f acc = {0.f, 0.f, 0.f, 0.f, 0.f, 0.f, 0.f, 0.f};
#pragma unroll
    for (int k0 = 0; k0 < DIM; k0 += 4) {
        v2f a, b;
        a.x = up[k0 + kh];
        a.y = up[k0 + kh + 1];
        b.x = ip[k0 + kh];
        b.y = ip[k0 + kh + 1];
        // (neg_a, A, neg_b, B, c_mod, C, reuse_a, reuse_b)
        acc = __builtin_amdgcn_wmma_f32_16x16x4_f32(
            false, a, false, b, (short)0, acc, false, false);
    }

    int   m    = -1;
    float pred = 0.0f;
    if (lane < 8)                     { m = lane;      pred = acc[lane]; }
    else if (lane >= 16 && lane < 24) { m = lane - 8;  pred = acc[lane - 16]; }
    if (m >= 0) {
        int o = tile * 16 + m;
        if (o < B) out[o] = pred * 0.25f;   // mean over (GCN_LAYERS+1)=4 layers
    }
}

// ---------------------------------------------------------------------------
extern "C" void kernel_launch(void* const* d_in, const int* in_sizes, int n_in,
                              void* d_out, int out_size, void* d_ws, size_t ws_size,
                              hipStream_t stream) {
    const int*   user_ids  = (const int*)d_in[0];
    const int*   item_ids  = (const int*)d_in[1];
    const int*   edge_user = (const int*)d_in[2];
    const int*   edge_item = (const int*)d_in[3];
    const float* user_emb  = (const float*)d_in[4];
    const float* item_emb  = (const float*)d_in[5];

    const int B       = in_sizes[0];
    const int E       = in_sizes[2];
    const int n_user  = in_sizes[4] / DIM;
    const int n_item  = in_sizes[5] / DIM;
    const int n_total = n_user + n_item;
    float* out = (float*)d_out;

    // Workspace carve-up (256B aligned): dinv | emb_a | emb_b | uacc | iacc
    char*  ws  = (char*)d_ws;
    size_t off = 0;
    auto carve = [&](size_t bytes) -> float* {
        float* p = (float*)(ws + off);
        off = (off + bytes + 255) & ~(size_t)255;
        return p;
    };
    float* deg   = carve((size_t)n_total * sizeof(float));
    float* emb_a = carve((size_t)n_total * DIM * sizeof(float));
    float* emb_b = carve((size_t)n_total * DIM * sizeof(float));
    float* uacc  = carve((size_t)B * DIM * sizeof(float));
    float* iacc  = carve((size_t)B * DIM * sizeof(float));

    // --- degree + normalization ---
    hipMemsetAsync(deg, 0, (size_t)n_total * sizeof(float), stream);
    k_degree<<<(E + 255) / 256, 256, 0, stream>>>(edge_user, edge_item, deg, E, n_user);
    k_rsqrt<<<(n_total + 255) / 256, 256, 0, stream>>>(deg, n_total);

    // --- layer-0 embeddings: ego = concat(user_emb, item_emb) ---
    hipMemcpyAsync(emb_a, user_emb, (size_t)n_user * DIM * sizeof(float),
                   hipMemcpyDeviceToDevice, stream);
    hipMemcpyAsync(emb_a + (size_t)n_user * DIM, item_emb,
                   (size_t)n_item * DIM * sizeof(float),
                   hipMemcpyDeviceToDevice, stream);

    {
        long long nt = (long long)B * 16;
        k_init_acc<<<(unsigned)((nt + 255) / 256), 256, 0, stream>>>(
            user_ids, item_ids, user_emb, item_emb, uacc, iacc, B);
    }

    // --- 3 LightGCN propagation layers ---
    float* cur = emb_a;
    float* nxt = emb_b;
    for (int layer = 0; layer < 3; ++layer) {
        hipMemsetAsync(nxt, 0, (size_t)n_total * DIM * sizeof(float), stream);
        long long nt = (long long)E * 16;
        k_spmm<<<(unsigned)((nt + 255) / 256), 256, 0, stream>>>(
            edge_user, edge_item, deg, cur, nxt, E, n_user);
        long long na = (long long)B * 16;
        k_acc<<<(unsigned)((na + 255) / 256), 256, 0, stream>>>(
            user_ids, item_ids, nxt, uacc, iacc, B, n_user);
        float* t = cur; cur = nxt; nxt = t;
    }

    // --- WMMA dot products: predictions = diag(U I^T) / 4 ---
    k_predict<<<(B + 15) / 16, 32, 0, stream>>>(uacc, iacc, out, B);
}